// FGA_18932215841221
// MI455X (gfx1250) — compile-verified
//
#include <hip/hip_runtime.h>
#include <math.h>
#include <cmath>

// Problem constants (fixed by the reference): B=8, C=1, H=W=2048, N_BAND=8.
#define HW   (2048 * 2048)
#define NB   8
#define NBAT 8

typedef float v2f __attribute__((ext_vector_type(2)));
typedef float v8f __attribute__((ext_vector_type(8)));

// Squared-distance thresholds for the radial bands, computed on the HOST in
// double precision and passed by value. band(d2) = #{k in 1..7 : d2 >= t[k]}.
// This is exactly floor(log1p(sqrt(d2)/1024*(e-1))*7) clamped to [0,7],
// since the band index is monotone in distance.
struct Thr { float t[8]; };   // t[0] unused

__device__ __forceinline__ int band_of_d2(float d2, const Thr& th) {
    int band = 0;
#pragma unroll
    for (int k = 1; k < 8; ++k) band += (d2 >= th.t[k]) ? 1 : 0;   // branchless
    return band;
}

// ---------------------------------------------------------------------------
// Pass 0: zero the 64 (batch, band) accumulators each launch (atomics follow).
__global__ void fga_zero(float* __restrict__ sums) {
    sums[threadIdx.x] = 0.0f;   // launched with 64 threads
}

// ---------------------------------------------------------------------------
// Pass 1: segment-sum of log1p(amp) into sums[batch*8 + band].
// Band from 7 compares on exact integer d2 (no sqrt/log). 32 replicated bin
// sets (one per 8-lane group) cap ds_add_f32 serialization at 8-way.
__global__ __launch_bounds__(256) void fga_reduce(const float* __restrict__ amp,
                                                  float* __restrict__ sums,
                                                  const Thr th) {
    __shared__ float sbins[32 * 64];          // 32 copies x (batch,band)
    const int tid = threadIdx.x;
#pragma unroll
    for (int i = tid; i < 32 * 64; i += 256) sbins[i] = 0.0f;
    __syncthreads();

    float* mybins = &sbins[(tid >> 3) * 64];  // per-8-lane-group copy
    const int base = blockIdx.x * 2048 + tid; // block covers 2048 pixels
#pragma unroll
    for (int i = 0; i < 8; ++i) {
        const int pix = base + i * 256;
        const int dy = (pix >> 11) - 1024;
        const int dx = (pix & 2047) - 1024;
        const float d2 = (float)(dy * dy + dx * dx);   // exact in fp32
        const int band = band_of_d2(d2, th);
#pragma unroll
        for (int b = 0; b < NBAT; ++b) {
            const float v = log1pf(amp[(size_t)b * HW + pix]);
            atomicAdd(&mybins[b * NB + band], v);      // ds_add_f32
        }
    }
    __syncthreads();
    if (tid < 64) {
        float acc = 0.0f;
#pragma unroll
        for (int c = 0; c < 32; ++c) acc += sbins[c * 64 + tid];
        atomicAdd(&sums[tid], acc);                    // 64 global atomics/block
    }
}

// ---------------------------------------------------------------------------
// Pass 2: the 8x8 MLP + gamma gate on ONE wave32 using V_WMMA_F32_16X16X4_F32.
//   layer1: e(8x8) @ w1^T(8x4)  -> two K-accumulating 16x16x4 WMMAs (K=8)
//   layer2: hdn(8x4) @ w2^T(4x8)-> one 16x16x4 WMMA (K=4)
// A-frag: lane%16 = M row, VGPR j holds K = 2*(lane/16)+j.
// C/D:    VGPR r  = M row r + 8*(lane/16), lane%16 = N col.
// All weight loads use clamped in-bounds indices + value masking (v_cndmask)
// instead of conditional loads, avoiding exec-mask branches.
__global__ __launch_bounds__(32) void fga_mlp(const float* __restrict__ sums,
                                              const float* __restrict__ g_base,
                                              const float* __restrict__ w1,
                                              const float* __restrict__ b1,
                                              const float* __restrict__ w2,
                                              const float* __restrict__ b2,
                                              float* __restrict__ gamma) {
    __shared__ float hsm[16 * 4];     // hdn in row-major [m][k]
    const int lane = threadIdx.x;
    const int half = lane >> 4;       // 0 or 1
    const int mr   = lane & 15;
    const int k0   = 2 * half;
    const int mr3  = mr & 3;          // clamped indices (always in-bounds)
    const int mr7  = mr & 7;
    const float m4 = (mr < 4) ? 1.0f : 0.0f;   // value masks (v_cndmask)
    const float m8 = (mr < 8) ? 1.0f : 0.0f;
    const float scale = 1.0f / (float)HW;

    // A fragments for layer 1: e[m][k] = sums[m*8+k]/(H*W), zero-padded rows
    v2f aLo, aHi, bLo, bHi;
    aLo.x = sums[mr7 * NB + k0]     * scale * m8;
    aLo.y = sums[mr7 * NB + k0 + 1] * scale * m8;
    aHi.x = sums[mr7 * NB + 4 + k0]     * scale * m8;
    aHi.y = sums[mr7 * NB + 4 + k0 + 1] * scale * m8;
    // B = w1^T: B[k=band][n=hid] = w1[hid*8 + band]; zero-pad n>=4
    bLo.x = w1[mr3 * NB + k0]         * m4;
    bLo.y = w1[mr3 * NB + k0 + 1]     * m4;
    bHi.x = w1[mr3 * NB + 4 + k0]     * m4;
    bHi.y = w1[mr3 * NB + 4 + k0 + 1] * m4;

    v8f c = {};
    c = __builtin_amdgcn_wmma_f32_16x16x4_f32(false, aLo, false, bLo, (short)0, c, false, false);
    c = __builtin_amdgcn_wmma_f32_16x16x4_f32(false, aHi, false, bHi, (short)0, c, false, false);

    // bias + exact GELU, then transpose C-layout -> row-major LDS for layer-2 A
    const float bias1 = b1[mr3];      // unconditional load
#pragma unroll
    for (int r = 0; r < 8; ++r) {
        const int m = r + 8 * half;
        const float x = c[r] + bias1 * m4;
        const float g = 0.5f * x * (1.0f + erff(x * 0.70710678118654752f));
        if (mr < 4) hsm[m * 4 + mr] = g;   // predicated store only
    }
    __syncthreads();

    // layer 2: A = hdn (K=4 exactly), B = w2^T: B[k=hid][n=band] = w2[band*4+hid]
    v2f a2, bf2;
    a2.x  = hsm[mr * 4 + k0];
    a2.y  = hsm[mr * 4 + k0 + 1];
    bf2.x = w2[mr7 * 4 + k0]     * m8;
    bf2.y = w2[mr7 * 4 + k0 + 1] * m8;

    v8f c2 = {};
    c2 = __builtin_amdgcn_wmma_f32_16x16x4_f32(false, a2, false, bf2, (short)0, c2, false, false);

    // sigmoid + softplus gate -> gamma_band[batch*8 + band]
    const float bias2 = b2[mr7];
    const float gb    = g_base[mr7];
    const float sp    = (gb > 20.0f) ? gb : log1pf(expf(gb));
#pragma unroll
    for (int r = 0; r < 8; ++r) {
        const int m = r + 8 * half;
        if (m < 8 && mr < 8) {
            const float x     = c2[r] + bias2;
            const float alpha = 1.0f / (1.0f + expf(-x));
            gamma[m * NB + mr] = 1.0f + sp * alpha;
        }
    }
}

// ---------------------------------------------------------------------------
// Pass 3: out = amp * gamma[b][band(h,w)]. float4 streaming; band (7 compares)
// computed once per (h,w) quad and reused across all 8 batch planes.
__global__ __launch_bounds__(256) void fga_apply(const float* __restrict__ amp,
                                                 const float* __restrict__ gamma,
                                                 float* __restrict__ out,
                                                 const Thr th) {
    __shared__ float sg[64];
    const int tid = threadIdx.x;
    if (tid < 64) sg[tid] = gamma[tid];
    __syncthreads();

    const int pixBase = (blockIdx.x * 256 + tid) * 4;
    const int dy = (pixBase >> 11) - 1024;        // same row for all 4 pixels
    const float dy2 = (float)(dy * dy);
    const int x0 = (pixBase & 2047) - 1024;
    int bd[4];
#pragma unroll
    for (int j = 0; j < 4; ++j) {
        const int dx = x0 + j;
        bd[j] = band_of_d2(dy2 + (float)(dx * dx), th);
    }

#pragma unroll
    for (int b = 0; b < NBAT; ++b) {
        const size_t off = (size_t)b * HW + pixBase;
        const float4 a = *(const float4*)(amp + off);
        float4 o;
        o.x = a.x * sg[b * NB + bd[0]];
        o.y = a.y * sg[b * NB + bd[1]];
        o.z = a.z * sg[b * NB + bd[2]];
        o.w = a.w * sg[b * NB + bd[3]];
        *(float4*)(out + off) = o;
    }
}

// ---------------------------------------------------------------------------
extern "C" void kernel_launch(void* const* d_in, const int* in_sizes, int n_in,
                              void* d_out, int out_size, void* d_ws, size_t ws_size,
                              hipStream_t stream) {
    const float* amp    = (const float*)d_in[0];   // (8,1,2048,2048) f32
    const float* g_base = (const float*)d_in[1];   // (8,)
    const float* w1     = (const float*)d_in[2];   // (4,8)
    const float* b1     = (const float*)d_in[3];   // (4,)
    const float* w2     = (const float*)d_in[4];   // (8,4)
    const float* b2     = (const float*)d_in[5];   // (8,)
    float* out   = (float*)d_out;
    float* sums  = (float*)d_ws;                   // 64 floats: (batch, band) sums
    float* gamma = sums + 64;                      // 64 floats: gamma_band

    // Band thresholds in squared-pixel-distance units, double precision on host:
    // band >= k  <=>  dist_norm >= (e^{k/7}-1)/(e-1)  <=>  d2 >= t[k]
    Thr th;
    th.t[0] = 0.0f;
    for (int k = 1; k < 8; ++k) {
        const double d = (std::exp((double)k / 7.0) - 1.0)
                       / 1.7182818284590452354 * 1024.0;
        th.t[k] = (float)(d * d);
    }

    fga_zero  <<<1,          64,  0, stream>>>(sums);
    fga_reduce<<<HW / 2048,  256, 0, stream>>>(amp, sums, th);
    fga_mlp   <<<1,          32,  0, stream>>>(sums, g_base, w1, b1, w2, b2, gamma);
    fga_apply <<<HW / 1024,  256, 0, stream>>>(amp, gamma, out, th);
}